// LSTMMeasurementSelector_39298950758819
// MI455X (gfx1250) — compile-verified
//
#include <hip/hip_runtime.h>
#include <math.h>

#define NQ 6
#define BS 3
#define HID 32
#define SNAPS 16
#define BATCH 1024
#define DIM 64
#define INDIM 24   // NQ + NQ*BS
#define G4 128     // 4*HID

typedef __attribute__((ext_vector_type(2))) float v2f;
typedef __attribute__((ext_vector_type(8))) float v8f;

__device__ __forceinline__ float sigmoidf_(float x) { return 1.0f / (1.0f + __expf(-x)); }

// D = A(16x4 f32) * B(4x16 f32) + C  -> v_wmma_f32_16x16x4_f32
__device__ __forceinline__ v8f wmma4_f32(v2f a, v2f b, v8f c) {
  return __builtin_amdgcn_wmma_f32_16x16x4_f32(false, a, false, b, (short)0, c, false, false);
}

// ---------------------------------------------------------------------------
// Kernel 1: 15-step LSTM + probs + snaps (= sum_s probs^2, via Pauli identity:
// (p.sigma)^2 = |p|^2 I, so every measured trace ratio is sum_s probs^2 and the
// rho evolution is unnecessary).
// One block = 16 batch rows, 8 waves. GEMMs via V_WMMA_F32_16X16X4_F32 with
// weight B-fragments hoisted into VGPRs (loop-invariant across the 15 steps).
// ---------------------------------------------------------------------------
__global__ __launch_bounds__(256) void k_lstm(
    const float* __restrict__ snapshot,   // [1024][6]
    const float* __restrict__ bcv,        // [1024][6][3]
    const float* __restrict__ h0,         // [1024][32]
    const float* __restrict__ c0,         // [1024][32]
    const float* __restrict__ W_ih,       // [128][24]
    const float* __restrict__ W_hh,       // [128][32]
    const float* __restrict__ b_ih,       // [128]
    const float* __restrict__ b_hh,       // [128]
    const float* __restrict__ Wp,         // [6][32][3]
    const float* __restrict__ bp,         // [6][3]
    float* __restrict__ feats,            // ws: [1024][16][24]
    float* __restrict__ lastbv)           // [1024][6][3]
{
  __shared__ float sWihT[INDIM][G4];
  __shared__ float sWhhT[HID][G4];
  __shared__ float sBias[G4];
  __shared__ float sWp[NQ * HID * BS];
  __shared__ float sBp[NQ * BS];
  __shared__ float sX[16][INDIM];
  __shared__ float sH[16][HID];
  __shared__ float sC[16][HID];
  __shared__ float sG[16][G4];
  __shared__ float sL[16][NQ * BS];

  const int tid  = threadIdx.x;
  const int lane = tid & 31;
  const int wave = tid >> 5;
  const int b0   = blockIdx.x * 16;

  // Stage weights in LDS (transposed for B-fragment access)
  for (int i = tid; i < G4 * INDIM; i += 256) { int n = i / INDIM, k = i % INDIM; sWihT[k][n] = W_ih[i]; }
  for (int i = tid; i < G4 * HID;   i += 256) { int n = i / HID,   k = i % HID;   sWhhT[k][n] = W_hh[i]; }
  for (int i = tid; i < G4;         i += 256) sBias[i] = b_ih[i] + b_hh[i];
  for (int i = tid; i < NQ*HID*BS;  i += 256) sWp[i] = Wp[i];
  for (int i = tid; i < NQ*BS;      i += 256) sBp[i] = bp[i];
  for (int i = tid; i < 16 * NQ;    i += 256) { int r = i / NQ, q = i % NQ; sX[r][q] = snapshot[(b0 + r) * NQ + q]; }
  for (int i = tid; i < 16 * 18;    i += 256) { int r = i / 18, j = i % 18; sX[r][NQ + j] = bcv[(b0 + r) * 18 + j]; }
  for (int i = tid; i < 16 * HID;   i += 256) {
    int r = i >> 5, j = i & 31;
    sH[r][j] = h0[(b0 + r) * HID + j];
    sC[r][j] = c0[(b0 + r) * HID + j];
  }
  __syncthreads();

  // feat 0 = [snapshot, basis_comp_vector]
  for (int i = tid; i < 16 * INDIM; i += 256) {
    int r = i / INDIM, j = i % INDIM;
    feats[((size_t)(b0 + r) * SNAPS + 0) * INDIM + j] = sX[r][j];
  }

  // Hoist loop-invariant weight B-fragments into registers (wave owns 16 cols)
  const int n    = wave * 16 + (lane & 15);
  const int koff = (lane >= 16) ? 2 : 0;
  const int row  = lane & 15;
  v2f wih[INDIM / 4];
  v2f whh[HID / 4];
  #pragma unroll
  for (int kc = 0; kc < INDIM / 4; ++kc) {
    wih[kc].x = sWihT[kc * 4 + koff][n];
    wih[kc].y = sWihT[kc * 4 + koff + 1][n];
  }
  #pragma unroll
  for (int kc = 0; kc < HID / 4; ++kc) {
    whh[kc].x = sWhhT[kc * 4 + koff][n];
    whh[kc].y = sWhhT[kc * 4 + koff + 1][n];
  }
  const float bias  = sBias[n];
  const int   rbase = (lane >= 16) ? 8 : 0;

  for (int step = 0; step < SNAPS - 1; ++step) {
    // ---- g = X @ W_ih^T + H @ W_hh^T + bias
    v8f acc = {0, 0, 0, 0, 0, 0, 0, 0};
    #pragma unroll
    for (int kc = 0; kc < INDIM / 4; ++kc) {
      v2f a; a.x = sX[row][kc * 4 + koff]; a.y = sX[row][kc * 4 + koff + 1];
      acc = wmma4_f32(a, wih[kc], acc);
    }
    #pragma unroll
    for (int kc = 0; kc < HID / 4; ++kc) {
      v2f a; a.x = sH[row][kc * 4 + koff]; a.y = sH[row][kc * 4 + koff + 1];
      acc = wmma4_f32(a, whh[kc], acc);
    }
    #pragma unroll
    for (int v = 0; v < 8; ++v) sG[rbase + v][n] = acc[v] + bias;
    __syncthreads();

    // ---- LSTM cell update
    for (int i = tid; i < 16 * HID; i += 256) {
      int r = i >> 5, j = i & 31;
      float gi = sG[r][j], gf = sG[r][32 + j], gg = sG[r][64 + j], go = sG[r][96 + j];
      float cc = sigmoidf_(gf) * sC[r][j] + sigmoidf_(gi) * tanhf(gg);
      float hh = sigmoidf_(go) * tanhf(cc);
      sC[r][j] = cc; sH[r][j] = hh;
    }
    __syncthreads();

    // ---- projection logits: h @ Wp + bp
    if (tid < 16 * NQ * BS) {
      int r = tid / 18; int j = tid % 18; int q = j / 3, s = j % 3;
      float l = sBp[j];
      const float* wp = &sWp[q * HID * BS + s];
      #pragma unroll
      for (int k = 0; k < HID; ++k) l += sH[r][k] * wp[k * BS];
      sL[r][j] = l;
    }
    __syncthreads();

    // ---- softmax over s, snaps[q] = sum_s probs^2  (measurement identity)
    if (tid < 16 * NQ) {
      int r = tid / NQ, q = tid % NQ;
      float l0 = sL[r][q*3], l1 = sL[r][q*3+1], l2 = sL[r][q*3+2];
      float mx = fmaxf(l0, fmaxf(l1, l2));
      float e0 = __expf(l0 - mx), e1 = __expf(l1 - mx), e2 = __expf(l2 - mx);
      float inv = 1.0f / (e0 + e1 + e2);
      float p0 = e0 * inv, p1 = e1 * inv, p2 = e2 * inv;
      sX[r][NQ + q*3]     = p0;
      sX[r][NQ + q*3 + 1] = p1;
      sX[r][NQ + q*3 + 2] = p2;
      sX[r][q] = p0*p0 + p1*p1 + p2*p2;
    }
    __syncthreads();

    for (int i = tid; i < 16 * INDIM; i += 256) {
      int r = i / INDIM, j = i % INDIM;
      feats[((size_t)(b0 + r) * SNAPS + step + 1) * INDIM + j] = sX[r][j];
    }
    if (step == SNAPS - 2) {
      for (int i = tid; i < 16 * 18; i += 256) {
        int r = i / 18, j = i % 18;
        lastbv[(b0 + r) * 18 + j] = sX[r][NQ + j];
      }
    }
  }
}

// ---------------------------------------------------------------------------
// Kernel 2: shadow reconstruction.
//   rc[t,q] = (I + 3*shadow*(bv.sigma))/2  (2x2 complex)
//   L[t] = kron(rc0,rc1,rc2) (8x8), R[t] = kron(rc3,rc4,rc5) (8x8)
//   OUT[i,j] = sum_t L[t,i]*R[t,j]   -> 64x64x16 complex GEMM via WMMA f32
//   out(r,c): r = (i>>3)*8 + (j>>3), c = (i&7)*8 + (j&7)
// Tiles scatter into LDS; final store is fully coalesced float4 streams.
// One block per batch element.
// ---------------------------------------------------------------------------
__global__ __launch_bounds__(256) void k_recon(
    const float* __restrict__ feats,  // [1024][16][24]
    float* __restrict__ out)          // [1024][2][64][64]
{
  __shared__ float sF[SNAPS][INDIM];
  __shared__ float rcr[SNAPS][NQ][4];
  __shared__ float rci[SNAPS][NQ][4];
  __shared__ float sLr[SNAPS][64];
  __shared__ float sLi[SNAPS][64];
  __shared__ float sRr[SNAPS][64];
  __shared__ float sRi[SNAPS][64];
  __shared__ float sOut[2][DIM][DIM];   // staged output (re, im)

  const int tid  = threadIdx.x;
  const int lane = tid & 31;
  const int wave = tid >> 5;
  const int b    = blockIdx.x;

  for (int i = tid; i < SNAPS * INDIM; i += 256)
    sF[i / INDIM][i % INDIM] = feats[(size_t)b * SNAPS * INDIM + i];
  __syncthreads();

  if (tid < SNAPS * NQ) {
    int t = tid / NQ, q = tid % NQ;
    float s  = sF[t][q];
    float bx = sF[t][NQ + q*3], by = sF[t][NQ + q*3 + 1], bz = sF[t][NQ + q*3 + 2];
    float a = 1.5f * s;   // 3*s/2
    rcr[t][q][0] = 0.5f + a * bz; rcr[t][q][1] = a * bx;
    rcr[t][q][2] = a * bx;        rcr[t][q][3] = 0.5f - a * bz;
    rci[t][q][0] = 0.0f;          rci[t][q][1] = -a * by;
    rci[t][q][2] = a * by;        rci[t][q][3] = 0.0f;
  }
  __syncthreads();

  // Build 8x8 kron halves per snapshot t
  for (int e = tid; e < SNAPS * 2 * 64; e += 256) {
    int t = e >> 7; int rem = e & 127; int side = rem >> 6; int idx = rem & 63;
    int rh = idx >> 3, ch = idx & 7;
    int q0 = side * 3;
    float pr = 1.0f, pi = 0.0f;
    #pragma unroll
    for (int qq = 0; qq < 3; ++qq) {        // qubit q0 is most-significant bit
      int rb = (rh >> (2 - qq)) & 1;
      int cb = (ch >> (2 - qq)) & 1;
      float xr = rcr[t][q0 + qq][rb * 2 + cb];
      float xi = rci[t][q0 + qq][rb * 2 + cb];
      float nr = pr * xr - pi * xi;
      float ni = pr * xi + pi * xr;
      pr = nr; pi = ni;
    }
    if (side == 0) { sLr[t][idx] = pr; sLi[t][idx] = pi; }
    else           { sRr[t][idx] = pr; sRi[t][idx] = pi; }
  }
  __syncthreads();

  // Complex GEMM  OUT = L^T R  (M=64, N=64, K=16), wave owns (part, m-tile)
  const int part  = wave & 1;    // 0=real, 1=imag
  const int m     = wave >> 1;   // m-tile 0..3
  const int koff  = (lane >= 16) ? 2 : 0;
  const int il    = m * 16 + (lane & 15);
  const int rbase = (lane >= 16) ? 8 : 0;

  for (int nt = 0; nt < 4; ++nt) {
    const int jl = nt * 16 + (lane & 15);
    v8f acc = {0, 0, 0, 0, 0, 0, 0, 0};
    #pragma unroll
    for (int kc = 0; kc < 4; ++kc) {
      int t0 = kc * 4 + koff, t1 = t0 + 1;
      v2f aLr; aLr.x = sLr[t0][il]; aLr.y = sLr[t1][il];
      v2f aLi; aLi.x = sLi[t0][il]; aLi.y = sLi[t1][il];
      v2f bRr; bRr.x = sRr[t0][jl]; bRr.y = sRr[t1][jl];
      v2f bRi; bRi.x = sRi[t0][jl]; bRi.y = sRi[t1][jl];
      if (part == 0) {
        acc = wmma4_f32(aLr, bRr, acc);          // + Lr^T Rr
        v2f aNeg; aNeg.x = -aLi.x; aNeg.y = -aLi.y;
        acc = wmma4_f32(aNeg, bRi, acc);         // - Li^T Ri
      } else {
        acc = wmma4_f32(aLr, bRi, acc);          // + Lr^T Ri
        acc = wmma4_f32(aLi, bRr, acc);          // + Li^T Rr
      }
    }
    // Scatter tile into LDS staging (bit-interleaved (i,j) -> (r,c))
    #pragma unroll
    for (int v = 0; v < 8; ++v) {
      int i = m * 16 + v + rbase;
      int j = jl;
      int r = ((i >> 3) << 3) | (j >> 3);
      int c = ((i & 7) << 3) | (j & 7);
      sOut[part][r][c] = acc[v] * 0.0625f;       // /SNAPS
    }
  }
  __syncthreads();

  // Fully coalesced streaming store: 2*64*64 floats as float4
  const float4* src4 = (const float4*)&sOut[0][0][0];
  float4* dst4 = (float4*)(out + (size_t)b * 2 * DIM * DIM);
  #pragma unroll
  for (int i = 0; i < (2 * DIM * DIM / 4) / 256; ++i)
    dst4[i * 256 + tid] = src4[i * 256 + tid];
}

// ---------------------------------------------------------------------------
extern "C" void kernel_launch(void* const* d_in, const int* in_sizes, int n_in,
                              void* d_out, int out_size, void* d_ws, size_t ws_size,
                              hipStream_t stream) {
  (void)in_sizes; (void)n_in; (void)out_size; (void)ws_size;
  const float* snapshot = (const float*)d_in[0];
  const float* bcv      = (const float*)d_in[1];
  // d_in[2] = rho — provably unused: (p.sigma)^2 = |p|^2 I makes every measured
  //   trace ratio equal sum_s probs^2, independent of rho (tr(rho)=1).
  const float* h0   = (const float*)d_in[3];
  const float* c0   = (const float*)d_in[4];
  const float* W_ih = (const float*)d_in[5];
  const float* W_hh = (const float*)d_in[6];
  const float* b_ih = (const float*)d_in[7];
  const float* b_hh = (const float*)d_in[8];
  const float* Wp   = (const float*)d_in[9];
  const float* bp   = (const float*)d_in[10];
  // d_in[11], d_in[12] = Pauli basis (baked into rc construction)

  float* out    = (float*)d_out;
  float* lastbv = out + (size_t)BATCH * 2 * DIM * DIM;
  float* feats  = (float*)d_ws;   // [1024][16][24] = 1.5 MB

  k_lstm <<<BATCH / 16, 256, 0, stream>>>(snapshot, bcv, h0, c0, W_ih, W_hh,
                                          b_ih, b_hh, Wp, bp, feats, lastbv);
  k_recon<<<BATCH,      256, 0, stream>>>(feats, out);
}